// Decoder_29222957482099
// MI455X (gfx1250) — compile-verified
//
#include <hip/hip_runtime.h>

typedef __attribute__((ext_vector_type(2))) float v2f;
typedef __attribute__((ext_vector_type(8))) float v8f;

namespace {
constexpr int B_    = 128;
constexpr int T_    = 200;
constexpr int V_    = 50257;
constexpr int H_    = 256;
constexpr int E_    = 128;
constexpr int OOV_  = 50;
constexpr int G3_   = 3 * H_;        // 768
constexpr int TWOH_ = 2 * H_;        // 512
constexpr int KIN_  = E_ + TWOH_;    // 640
constexpr int OUTROW_ = V_ + OOV_;   // 50307

// d_out layout (floats): prob_out | decode_hidden | selective_read_new
constexpr size_t OUT_HID = (size_t)B_ * OUTROW_;
constexpr size_t OUT_SEL = OUT_HID + (size_t)B_ * H_;

// workspace layout (floats)
constexpr size_t WS_GRUIN = 0;                            // B*640
constexpr size_t WS_GI    = WS_GRUIN + (size_t)B_ * KIN_; // B*768
constexpr size_t WS_GH    = WS_GI + (size_t)B_ * G3_;     // B*768
constexpr size_t WS_SC    = WS_GH + (size_t)B_ * G3_;     // B*T (score_c accumulator)
constexpr size_t WS_PC    = WS_SC + (size_t)B_ * T_;      // B*T (normalized prob_c)
} // namespace

__device__ __forceinline__ v8f wmma_f32_k4(v2f a, v2f b, v8f c) {
  // V_WMMA_F32_16X16X4_F32 : true fp32 matrix op (CDNA5)
  return __builtin_amdgcn_wmma_f32_16x16x4_f32(false, a, false, b, (short)0, c,
                                               false, false);
}

// Build gru_in = [selective_read | emb[dec]] ; zero score_c accumulator
__global__ void k_prep(const float* __restrict__ selr,
                       const float* __restrict__ emb,
                       const int* __restrict__ dec,
                       float* __restrict__ gruin, float* __restrict__ sc) {
  int i = blockIdx.x * blockDim.x + threadIdx.x;
  if (i < B_ * KIN_) {
    int b = i / KIN_, k = i % KIN_;
    float v = (k < TWOH_) ? selr[(size_t)b * TWOH_ + k]
                          : emb[(size_t)dec[b] * E_ + (k - TWOH_)];
    gruin[i] = v;
  } else if (i < B_ * (KIN_ + T_)) {
    sc[i - B_ * KIN_] = 0.0f;
  }
}

// C[MxN] = A[MxK] @ W[NxK]^T + bias
// Register-blocked WMMA fp32: one wave computes a 32x64 macro-tile
// (2 M-frags x 4 N-frags -> 8 accumulators; 6 loads feed 8 WMMAs per K-step).
__global__ void __launch_bounds__(32)
k_gemm_xwt(const float* __restrict__ A, const float* __restrict__ W,
           const float* __restrict__ bias, float* __restrict__ C,
           int M, int N, int K, int lda, int ldw, int ldc) {
  const int lane = threadIdx.x;
  const int half = lane >> 4;     // 0: K={0,1}, 1: K={2,3} within k-step
  const int l16  = lane & 15;
  const int tm = blockIdx.x * 32;
  const int tn = blockIdx.y * 64;
  const int kh = 2 * half;

  const float* __restrict__ arow0 = A + (size_t)min(tm + l16,      M - 1) * lda;
  const float* __restrict__ arow1 = A + (size_t)min(tm + 16 + l16, M - 1) * lda;
  const float* __restrict__ wrow0 = W + (size_t)min(tn +  0 + l16, N - 1) * ldw;
  const float* __restrict__ wrow1 = W + (size_t)min(tn + 16 + l16, N - 1) * ldw;
  const float* __restrict__ wrow2 = W + (size_t)min(tn + 32 + l16, N - 1) * ldw;
  const float* __restrict__ wrow3 = W + (size_t)min(tn + 48 + l16, N - 1) * ldw;

  v8f acc[2][4] = {};
  for (int k0 = 0; k0 < K; k0 += 4) {
    v2f a0 = *(const v2f*)(arow0 + k0 + kh);
    v2f a1 = *(const v2f*)(arow1 + k0 + kh);
    v2f b0 = *(const v2f*)(wrow0 + k0 + kh);
    v2f b1 = *(const v2f*)(wrow1 + k0 + kh);
    v2f b2 = *(const v2f*)(wrow2 + k0 + kh);
    v2f b3 = *(const v2f*)(wrow3 + k0 + kh);
    acc[0][0] = wmma_f32_k4(a0, b0, acc[0][0]);
    acc[0][1] = wmma_f32_k4(a0, b1, acc[0][1]);
    acc[0][2] = wmma_f32_k4(a0, b2, acc[0][2]);
    acc[0][3] = wmma_f32_k4(a0, b3, acc[0][3]);
    acc[1][0] = wmma_f32_k4(a1, b0, acc[1][0]);
    acc[1][1] = wmma_f32_k4(a1, b1, acc[1][1]);
    acc[1][2] = wmma_f32_k4(a1, b2, acc[1][2]);
    acc[1][3] = wmma_f32_k4(a1, b3, acc[1][3]);
  }

#pragma unroll
  for (int j = 0; j < 4; ++j) {
    const int n = tn + j * 16 + l16;
    if (n < N) {
      const float bn = bias[n];
#pragma unroll
      for (int i = 0; i < 2; ++i) {
#pragma unroll
        for (int v = 0; v < 8; ++v) {
          int m = tm + i * 16 + v + half * 8;
          if (m < M) C[(size_t)m * ldc + n] = acc[i][j][v] + bn;
        }
      }
    }
  }
}

// Fused: enc_proj = tanh(enc @ Wc^T + Wc_b); score_c_acc += enc_proj . hidden
// Same 32x64 register blocking; enc_proj never touches memory.
__global__ void __launch_bounds__(32)
k_encproj(const float* __restrict__ enc, const float* __restrict__ Wc,
          const float* __restrict__ Wcb, const float* __restrict__ hid,
          float* __restrict__ sc) {
  const int lane = threadIdx.x;
  const int half = lane >> 4;
  const int l16  = lane & 15;
  const int tm = blockIdx.x * 32;  // flattened (b*T + t) rows
  const int tn = blockIdx.y * 64;  // h block (N = 256 -> 4 blocks)
  const int kh = 2 * half;

  const float* __restrict__ arow0 = enc + (size_t)(tm + l16) * TWOH_;
  const float* __restrict__ arow1 = enc + (size_t)(tm + 16 + l16) * TWOH_;
  const float* __restrict__ wrow0 = Wc + (size_t)(tn +  0 + l16) * TWOH_;
  const float* __restrict__ wrow1 = Wc + (size_t)(tn + 16 + l16) * TWOH_;
  const float* __restrict__ wrow2 = Wc + (size_t)(tn + 32 + l16) * TWOH_;
  const float* __restrict__ wrow3 = Wc + (size_t)(tn + 48 + l16) * TWOH_;

  v8f acc[2][4] = {};
  for (int k0 = 0; k0 < TWOH_; k0 += 4) {
    v2f a0 = *(const v2f*)(arow0 + k0 + kh);
    v2f a1 = *(const v2f*)(arow1 + k0 + kh);
    v2f b0 = *(const v2f*)(wrow0 + k0 + kh);
    v2f b1 = *(const v2f*)(wrow1 + k0 + kh);
    v2f b2 = *(const v2f*)(wrow2 + k0 + kh);
    v2f b3 = *(const v2f*)(wrow3 + k0 + kh);
    acc[0][0] = wmma_f32_k4(a0, b0, acc[0][0]);
    acc[0][1] = wmma_f32_k4(a0, b1, acc[0][1]);
    acc[0][2] = wmma_f32_k4(a0, b2, acc[0][2]);
    acc[0][3] = wmma_f32_k4(a0, b3, acc[0][3]);
    acc[1][0] = wmma_f32_k4(a1, b0, acc[1][0]);
    acc[1][1] = wmma_f32_k4(a1, b1, acc[1][1]);
    acc[1][2] = wmma_f32_k4(a1, b2, acc[1][2]);
    acc[1][3] = wmma_f32_k4(a1, b3, acc[1][3]);
  }

  const int hbase = tn + l16;      // lane's h column within each j-frag (+16j)
  float bn[4];
#pragma unroll
  for (int j = 0; j < 4; ++j) bn[j] = Wcb[hbase + j * 16];

#pragma unroll
  for (int i = 0; i < 2; ++i) {
#pragma unroll
    for (int v = 0; v < 8; ++v) {
      const int m = tm + i * 16 + v + half * 8;  // global row in B*T
      const int b = m / T_, t = m - b * T_;
      float val = 0.0f;
#pragma unroll
      for (int j = 0; j < 4; ++j) {
        val += tanhf(acc[i][j][v] + bn[j]) * hid[(size_t)b * H_ + hbase + j * 16];
      }
#pragma unroll
      for (int s = 1; s < 16; s <<= 1) val += __shfl_xor(val, s, 32);
      if (l16 == 0) atomicAdd(&sc[b * T_ + t], val);
    }
  }
}

// GRU gates -> decode_hidden (written directly into d_out)
__global__ void k_gates(const float* __restrict__ gi, const float* __restrict__ gh,
                        const float* __restrict__ hprev, float* __restrict__ hid) {
  int i = blockIdx.x * blockDim.x + threadIdx.x;
  if (i >= B_ * H_) return;
  int b = i / H_, h = i % H_;
  const float* gib = gi + (size_t)b * G3_;
  const float* ghb = gh + (size_t)b * G3_;
  float r = 1.0f / (1.0f + __expf(-(gib[h] + ghb[h])));
  float z = 1.0f / (1.0f + __expf(-(gib[H_ + h] + ghb[H_ + h])));
  float n = tanhf(gib[2 * H_ + h] + r * ghb[2 * H_ + h]);
  hid[i] = (1.0f - z) * n + z * hprev[i];
}

// score_c = tanh(acc) + mask
__global__ void k_scmask(float* __restrict__ sc, const int* __restrict__ idxs) {
  int i = blockIdx.x * blockDim.x + threadIdx.x;
  if (i >= B_ * T_) return;
  sc[i] = tanhf(sc[i]) + ((idxs[i] == 0) ? -10000.0f : 0.0f);
}

// Joint softmax over [score_g (in d_out, len V) | score_c (len T)] per row.
// Writes normalized prob_g in place, prob_c to ws, fills OOV tail = 1e-4.
__global__ void __launch_bounds__(256)
k_softmax(float* __restrict__ pg, float* __restrict__ sc, float* __restrict__ pc) {
  __shared__ float red[256];
  const int b = blockIdx.x, tid = threadIdx.x;
  float* rowg = pg + (size_t)b * OUTROW_;
  float* rowc = sc + (size_t)b * T_;
  float mx = -3.0e38f;
  for (int i = tid; i < V_; i += 256) mx = fmaxf(mx, rowg[i]);
  for (int i = tid; i < T_; i += 256) mx = fmaxf(mx, rowc[i]);
  red[tid] = mx; __syncthreads();
  for (int s = 128; s > 0; s >>= 1) {
    if (tid < s) red[tid] = fmaxf(red[tid], red[tid + s]);
    __syncthreads();
  }
  mx = red[0]; __syncthreads();
  float sum = 0.0f;
  for (int i = tid; i < V_; i += 256) { float e = __expf(rowg[i] - mx); rowg[i] = e; sum += e; }
  for (int i = tid; i < T_; i += 256) { float e = __expf(rowc[i] - mx); rowc[i] = e; sum += e; }
  red[tid] = sum; __syncthreads();
  for (int s = 128; s > 0; s >>= 1) {
    if (tid < s) red[tid] += red[tid + s];
    __syncthreads();
  }
  const float inv = 1.0f / red[0];
  for (int i = tid; i < V_; i += 256) rowg[i] *= inv;
  for (int i = tid; i < T_; i += 256) pc[(size_t)b * T_ + i] = rowc[i] * inv;
  for (int i = V_ + tid; i < OUTROW_; i += 256) rowg[i] = 1.0e-4f;
}

// Copy-mechanism scatter: prob_out[b, idx[b,t]] += prob_c[b,t]
__global__ void k_scatter(const float* __restrict__ pc, const int* __restrict__ idxs,
                          float* __restrict__ prob) {
  int i = blockIdx.x * blockDim.x + threadIdx.x;
  if (i >= B_ * T_) return;
  int b = i / T_;
  atomicAdd(&prob[(size_t)b * OUTROW_ + idxs[i]], pc[i]);
}

// selective_read_new[b,e] = sum_t prob_c*match (normalized) * enc[b,t,e]
__global__ void __launch_bounds__(512)
k_selread(const int* __restrict__ idxs, const int* __restrict__ dec,
          const float* __restrict__ pc, const float* __restrict__ enc,
          float* __restrict__ out) {
  __shared__ float w[T_];
  __shared__ float red[512];
  const int b = blockIdx.x, tid = threadIdx.x;
  float m = 0.0f;
  if (tid < T_) m = (idxs[b * T_ + tid] == dec[b]) ? 1.0f : 0.0f;
  red[tid] = m; __syncthreads();
  for (int s = 256; s > 0; s >>= 1) {
    if (tid < s) red[tid] += red[tid + s];
    __syncthreads();
  }
  const float tot = red[0];
  const float scale = (tot > 1.0f) ? (1.0f / tot) : 1.0f;
  if (tid < T_) w[tid] = m * scale * pc[(size_t)b * T_ + tid];
  __syncthreads();
  float acc = 0.0f;
  const float* eb = enc + (size_t)b * T_ * TWOH_ + tid;  // e = tid (0..511)
  for (int t = 0; t < T_; ++t) acc += w[t] * eb[(size_t)t * TWOH_];
  out[(size_t)b * TWOH_ + tid] = acc;
}

extern "C" void kernel_launch(void* const* d_in, const int* in_sizes, int n_in,
                              void* d_out, int out_size, void* d_ws, size_t ws_size,
                              hipStream_t stream) {
  (void)in_sizes; (void)n_in; (void)out_size; (void)ws_size;
  const int*   dec   = (const int*)  d_in[0];
  const float* enc   = (const float*)d_in[1];
  const int*   idxs  = (const int*)  d_in[2];
  const float* hprev = (const float*)d_in[3];   // (1,B,H) contiguous
  const float* selr  = (const float*)d_in[4];
  // d_in[5] = step (==1 path implemented); d_in[11]/[12] (Wi_w/Wi_b) unused at step!=0
  const float* emb   = (const float*)d_in[6];
  const float* W_ih  = (const float*)d_in[7];
  const float* W_hh  = (const float*)d_in[8];
  const float* b_ih  = (const float*)d_in[9];
  const float* b_hh  = (const float*)d_in[10];
  const float* Wg_w  = (const float*)d_in[13];
  const float* Wg_b  = (const float*)d_in[14];
  const float* Wc_w  = (const float*)d_in[15];
  const float* Wc_b  = (const float*)d_in[16];

  float* out  = (float*)d_out;
  float* ws   = (float*)d_ws;
  float* gruin = ws + WS_GRUIN;
  float* gi    = ws + WS_GI;
  float* gh    = ws + WS_GH;
  float* sc    = ws + WS_SC;
  float* pc    = ws + WS_PC;
  float* prob  = out;             // score_g / prob_out live here
  float* hid   = out + OUT_HID;   // decode_hidden output
  float* sel   = out + OUT_SEL;   // selective_read output

  const int prepTot = B_ * (KIN_ + T_);
  k_prep<<<(prepTot + 255) / 256, 256, 0, stream>>>(selr, emb, dec, gruin, sc);

  // gi = gru_in @ W_ih^T + b_ih   (128x640x768), 32x64 tiles
  k_gemm_xwt<<<dim3(B_ / 32, G3_ / 64), 32, 0, stream>>>(
      gruin, W_ih, b_ih, gi, B_, G3_, KIN_, KIN_, KIN_, G3_);
  // gh = h @ W_hh^T + b_hh        (128x256x768)
  k_gemm_xwt<<<dim3(B_ / 32, G3_ / 64), 32, 0, stream>>>(
      hprev, W_hh, b_hh, gh, B_, G3_, H_, H_, H_, G3_);

  k_gates<<<(B_ * H_ + 255) / 256, 256, 0, stream>>>(gi, gh, hprev, hid);

  // fused enc_proj + score_c accumulation (25600x512x256)
  k_encproj<<<dim3((B_ * T_) / 32, H_ / 64), 32, 0, stream>>>(enc, Wc_w, Wc_b, hid, sc);

  // score_g straight into prob_out region (128x256x50257)
  k_gemm_xwt<<<dim3(B_ / 32, (V_ + 63) / 64), 32, 0, stream>>>(
      hid, Wg_w, Wg_b, prob, B_, V_, H_, H_, H_, OUTROW_);

  k_scmask<<<(B_ * T_ + 255) / 256, 256, 0, stream>>>(sc, idxs);
  k_softmax<<<B_, 256, 0, stream>>>(prob, sc, pc);
  k_scatter<<<(B_ * T_ + 255) / 256, 256, 0, stream>>>(pc, idxs, prob);
  k_selread<<<B_, 512, 0, stream>>>(idxs, dec, pc, enc, sel);
}